// SAGEDecoder_32959579030042
// MI455X (gfx1250) — compile-verified
//
#include <hip/hip_runtime.h>
#include <hip/hip_bf16.h>

typedef __attribute__((ext_vector_type(2))) float v2f;
typedef __attribute__((ext_vector_type(8))) float v8f;

#define IN_CH 128
#define HID_CH 256
#define OUT_CH 128

// ---------------------------------------------------------------------------
// Utility: zero a float buffer
// ---------------------------------------------------------------------------
__global__ void zero_kernel(float* __restrict__ p, size_t n) {
    size_t i = (size_t)blockIdx.x * blockDim.x + threadIdx.x;
    if (i < n) p[i] = 0.0f;
}

// ---------------------------------------------------------------------------
// Degree count: one thread per edge, f32 atomic into deg[dst]
// ---------------------------------------------------------------------------
__global__ void deg_kernel(const long long* __restrict__ dst,
                           float* __restrict__ deg, int n_edges) {
    int e = blockIdx.x * blockDim.x + threadIdx.x;
    if (e < n_edges) {
        atomicAdd(&deg[(int)dst[e]], 1.0f);
    }
}

__global__ void invdeg_kernel(float* __restrict__ deg, int n) {
    int i = blockIdx.x * blockDim.x + threadIdx.x;
    if (i < n) deg[i] = 1.0f / fmaxf(deg[i], 1.0f);
}

// ---------------------------------------------------------------------------
// Edge scatter-add: acc[dst] += feat[src].  C channels, 4 per thread (float4
// gather + 4 global_atomic_add_f32).  C is 128 or 256 -> TPE is a power of two
// so the / and % lower to shifts.
// ---------------------------------------------------------------------------
template <int C>
__global__ void scatter_feat(const float* __restrict__ feat,
                             const long long* __restrict__ src,
                             const long long* __restrict__ dst,
                             float* __restrict__ acc, int n_edges) {
    constexpr int TPE = C / 4;  // threads per edge
    long long idx = (long long)blockIdx.x * blockDim.x + threadIdx.x;
    long long e = idx / TPE;
    int ch = (int)(idx % TPE) * 4;
    if (e >= n_edges) return;
    int s = (int)src[e];
    int d = (int)dst[e];
    const float4 v = *(const float4*)(feat + (size_t)s * C + ch);
    float* o = acc + (size_t)d * C + ch;
    atomicAdd(o + 0, v.x);
    atomicAdd(o + 1, v.y);
    atomicAdd(o + 2, v.z);
    atomicAdd(o + 3, v.w);
}

// ---------------------------------------------------------------------------
// Fused SAGE GEMM strip kernel (fp32 WMMA):
//   out[16 x CT*16 strip] = (acc*inv_deg) @ w_l^T + x @ w_r^T + bias (opt ReLU)
//
// One wave (32 lanes) computes a 16(node) x CT*16(out-ch) strip: the A
// fragment is loaded ONCE per k-step and reused across CT column tiles
// (CT independent v8f accumulators -> no WMMA->WMMA RAW chain, 4x less A
// traffic from L2).
//   grid.x = n_nodes/16, grid.y = out_ch/(16*CT), block = 32 threads.
//
// Fragment layouts (ISA 7.12.2, 32-bit):
//   A 16x4: lane L holds row M=L&15; VGPR pair holds K = k+2*(L>>4), +1.
//   B 4x16: lane L holds column N=L&15; VGPR pair holds same K pair.
//           B[k][n] = W[n][k]  (weights are [out_ch, K] row-major).
//   C/D 16x16: VGPR v <-> M = v + 8*(L>>4), N = L&15.
// EXEC is all ones throughout (no divergence) as WMMA requires.
// ---------------------------------------------------------------------------
template <int K, int CT, bool RELU>
__global__ void sage_gemm_wmma(const float* __restrict__ acc,      // [N,K] msg sums
                               const float* __restrict__ inv_deg,  // [N]
                               const float* __restrict__ xin,      // [N,K]
                               const float* __restrict__ w_l,      // [out_ch,K]
                               const float* __restrict__ w_r,      // [out_ch,K]
                               const float* __restrict__ bias,     // [out_ch]
                               float* __restrict__ out,            // [N,out_ch]
                               int out_ch) {
    const int lane = threadIdx.x & 31;
    const int half = lane >> 4;       // 0: lanes 0-15, 1: lanes 16-31
    const int l15  = lane & 15;
    const int nodeTile = blockIdx.x;
    const int colBase  = blockIdx.y * (CT * 16);

    const int row = nodeTile * 16 + l15;  // node whose A row this lane feeds
    const int bk  = half * 2;             // K sub-offset inside the 4-wide step

    const float s = inv_deg[row];
    const float* aRow = acc + (size_t)row * K;
    const float* xRow = xin + (size_t)row * K;

    const float* wlCol[CT];
    const float* wrCol[CT];
#pragma unroll
    for (int t = 0; t < CT; ++t) {
        int col = colBase + t * 16 + l15;  // out channel this lane feeds for tile t
        wlCol[t] = w_l + (size_t)col * K;
        wrCol[t] = w_r + (size_t)col * K;
    }

    v8f c[CT];
#pragma unroll
    for (int t = 0; t < CT; ++t) c[t] = (v8f){};

    // term 1: mean-aggregated neighbors @ w_l^T  (scale fused into A load)
#pragma unroll 4
    for (int k = 0; k < K; k += 4) {
        v2f a;
        a.x = aRow[k + bk] * s;
        a.y = aRow[k + bk + 1] * s;
#pragma unroll
        for (int t = 0; t < CT; ++t) {
            v2f b;
            b.x = wlCol[t][k + bk];
            b.y = wlCol[t][k + bk + 1];
            c[t] = __builtin_amdgcn_wmma_f32_16x16x4_f32(false, a, false, b,
                                                         (short)0, c[t],
                                                         false, false);
        }
    }
    // term 2: self features @ w_r^T
#pragma unroll 4
    for (int k = 0; k < K; k += 4) {
        v2f a;
        a.x = xRow[k + bk];
        a.y = xRow[k + bk + 1];
#pragma unroll
        for (int t = 0; t < CT; ++t) {
            v2f b;
            b.x = wrCol[t][k + bk];
            b.y = wrCol[t][k + bk + 1];
            c[t] = __builtin_amdgcn_wmma_f32_16x16x4_f32(false, a, false, b,
                                                         (short)0, c[t],
                                                         false, false);
        }
    }

    // epilogue: bias (+ReLU) fused into the store
#pragma unroll
    for (int t = 0; t < CT; ++t) {
        int col = colBase + t * 16 + l15;
        const float bv = bias[col];
#pragma unroll
        for (int v = 0; v < 8; ++v) {
            int node = nodeTile * 16 + v + 8 * half;
            float val = c[t][v] + bv;
            if (RELU) val = fmaxf(val, 0.0f);
            out[(size_t)node * out_ch + col] = val;
        }
    }
}

// ---------------------------------------------------------------------------
// Launch
// ---------------------------------------------------------------------------
extern "C" void kernel_launch(void* const* d_in, const int* in_sizes, int n_in,
                              void* d_out, int out_size, void* d_ws, size_t ws_size,
                              hipStream_t stream) {
    const float*     x    = (const float*)d_in[0];
    const long long* ei   = (const long long*)d_in[1];  // int64 [2, E]
    const float*     w_l3 = (const float*)d_in[2];
    const float*     b_l3 = (const float*)d_in[3];
    const float*     w_r3 = (const float*)d_in[4];
    const float*     w_l4 = (const float*)d_in[5];
    const float*     b_l4 = (const float*)d_in[6];
    const float*     w_r4 = (const float*)d_in[7];
    float*           out  = (float*)d_out;

    const int N = in_sizes[0] / IN_CH;       // 50000
    const int E = in_sizes[1] / 2;           // 800000
    const long long* src = ei;
    const long long* dst = ei + E;

    // workspace layout: deg/inv_deg [N] | acc [N,256] | h [N,256]
    char* ws = (char*)d_ws;
    size_t degBytes = ((size_t)N * sizeof(float) + 255) & ~(size_t)255;
    float* deg = (float*)ws;
    float* acc = (float*)(ws + degBytes);
    float* h   = acc + (size_t)N * HID_CH;

    const int ZB = 256;
    // ---- degrees (shared by both layers) ----
    zero_kernel<<<(N + ZB - 1) / ZB, ZB, 0, stream>>>(deg, (size_t)N);
    deg_kernel<<<(E + ZB - 1) / ZB, ZB, 0, stream>>>(dst, deg, E);
    invdeg_kernel<<<(N + ZB - 1) / ZB, ZB, 0, stream>>>(deg, N);

    // ---- layer 3: 128 -> 256, ReLU ----
    {
        size_t accN = (size_t)N * IN_CH;
        zero_kernel<<<(int)((accN + ZB - 1) / ZB), ZB, 0, stream>>>(acc, accN);
        long long threads = (long long)E * (IN_CH / 4);
        scatter_feat<IN_CH><<<(int)((threads + ZB - 1) / ZB), ZB, 0, stream>>>(
            x, src, dst, acc, E);
        dim3 grid(N / 16, HID_CH / (16 * 4));
        sage_gemm_wmma<IN_CH, 4, true><<<grid, 32, 0, stream>>>(
            acc, deg, x, w_l3, w_r3, b_l3, h, HID_CH);
    }

    // ---- layer 4: 256 -> 128, no activation ----
    {
        size_t accN = (size_t)N * HID_CH;
        zero_kernel<<<(int)((accN + ZB - 1) / ZB), ZB, 0, stream>>>(acc, accN);
        long long threads = (long long)E * (HID_CH / 4);
        scatter_feat<HID_CH><<<(int)((threads + ZB - 1) / ZB), ZB, 0, stream>>>(
            h, src, dst, acc, E);
        dim3 grid(N / 16, OUT_CH / (16 * 4));
        sage_gemm_wmma<HID_CH, 4, false><<<grid, 32, 0, stream>>>(
            acc, deg, h, w_l4, w_r4, b_l4, out, OUT_CH);
    }
}